// S4Model_64364379897894
// MI455X (gfx1250) — compile-verified
//
#include <hip/hip_runtime.h>

// y = u @ W^T with W = C (I-A)^{-1} B + D  — the whole S4 reference collapses
// to one constant 128x128 matrix applied to 262144 rows.
// Big GEMM is HBM-bound (~256 MB @ 23.3 TB/s ~ 11us); f32 WMMA 16x16x4 keeps
// reference precision while putting the MACs on the matrix pipe.

typedef __attribute__((ext_vector_type(2))) float v2f;
typedef __attribute__((ext_vector_type(8))) float v8f;

#define S 128
#define LDS_STRIDE 132   // pad: 132 mod 64 = 4 -> conflict-free b64 fragment reads

// ---------------------------------------------------------------------------
// Kernel 1: W = C * (I-A)^{-1} * B + D   (single block, tiny)
// ---------------------------------------------------------------------------
__global__ __launch_bounds__(256) void s4_precompute_W(
    const float* __restrict__ A, const float* __restrict__ B,
    const float* __restrict__ C, const float* __restrict__ D,
    float* __restrict__ W)
{
    extern __shared__ float sm[];
    float* sA = sm;            // holds A, later reused for T = Minv*B
    float* sM = sm + S * S;    // holds Minv = (I-A)^{-1}
    const int tid = threadIdx.x;

    for (int i = tid; i < S * S; i += 256) sA[i] = A[i];
    __syncthreads();

    // Forward substitution: (I-A) M = I.  Column j is fully independent:
    // M[i][j] = (delta_ij + sum_{k<i} A[i][k] M[k][j]) / (1 - A[i][i])
    if (tid < S) {
        const int j = tid;
        for (int i = 0; i < S; ++i) {
            float s = (i == j) ? 1.0f : 0.0f;
            for (int k = 0; k < i; ++k) s += sA[i * S + k] * sM[k * S + j];
            sM[i * S + j] = s / (1.0f - sA[i * S + i]);
        }
    }
    __syncthreads();

    const int col = tid & (S - 1);
    const int r0  = (tid >> 7) * (S / 2);

    // T = Minv * B  (overwrite sA; A is dead)
    for (int s = r0; s < r0 + S / 2; ++s) {
        float acc = 0.0f;
        for (int k = 0; k < S; ++k) acc += sM[s * S + k] * B[k * S + col];
        sA[s * S + col] = acc;
    }
    __syncthreads();

    // W = C * T + D
    for (int o = r0; o < r0 + S / 2; ++o) {
        float acc = D[o * S + col];
        for (int s = 0; s < S; ++s) acc += C[o * S + s] * sA[s * S + col];
        W[o * S + col] = acc;
    }
}

// ---------------------------------------------------------------------------
// Kernel 2: y[M,128] = u[M,128] @ W^T via V_WMMA_F32_16X16X4_F32
// Block: 256 threads = 8 waves; each block does 128 rows (16 per wave),
// each wave accumulates all 8 N-tiles (8 x v8f accumulators).
// ---------------------------------------------------------------------------
__global__ __launch_bounds__(256) void s4_gemm_wmma(
    const float* __restrict__ u, const float* __restrict__ Wm,
    float* __restrict__ y)
{
    extern __shared__ float sm[];
    float* su = sm;                      // 128 x LDS_STRIDE (u chunk)
    float* sw = sm + S * LDS_STRIDE;     // 128 x LDS_STRIDE (W)

    const int tid  = threadIdx.x;
    const long base = (long)blockIdx.x * 128;

    // Coalesced b128 staging of W and this block's u rows.
    for (int i = tid; i < S * (S / 4); i += 256) {
        const int row = i >> 5;      // 32 float4 per row
        const int c4  = (i & 31) * 4;
        *(float4*)&sw[row * LDS_STRIDE + c4] =
            *(const float4*)&Wm[row * S + c4];
        *(float4*)&su[row * LDS_STRIDE + c4] =
            *(const float4*)&u[(base + row) * S + c4];
    }
    __syncthreads();

    const int wave = tid >> 5;
    const int lane = tid & 31;
    const int r    = lane & 15;      // M (for A) / N (for B) index within tile
    const int half = lane >> 4;      // selects K pair {0,1} vs {2,3}

    v8f acc[8];
    #pragma unroll
    for (int t = 0; t < 8; ++t) acc[t] = (v8f){};

    const float* aRow = &su[(16 * wave + r) * LDS_STRIDE];
    const int coff = 2 * half;

    for (int kk = 0; kk < 32; ++kk) {
        const int c = 4 * kk + coff;
        // A fragment: lane holds u[m0+r][c], u[m0+r][c+1]   (ISA 16x4 layout)
        const v2f a = *(const v2f*)(aRow + c);
        #pragma unroll
        for (int t = 0; t < 8; ++t) {
            // B fragment: Bmat[k][n] = W[16t + n][k]; lane holds K = c, c+1 at N = r
            const v2f b = *(const v2f*)(&sw[(16 * t + r) * LDS_STRIDE + c]);
            acc[t] = __builtin_amdgcn_wmma_f32_16x16x4_f32(
                false, a, false, b, (short)0, acc[t], false, false);
        }
    }

    // C/D layout: VGPR v, lane -> (M = v + 8*half, N = r)
    const long orow = base + 16 * wave + 8 * half;
    #pragma unroll
    for (int t = 0; t < 8; ++t) {
        #pragma unroll
        for (int v = 0; v < 8; ++v) {
            y[(orow + v) * S + 16 * t + r] = acc[t][v];
        }
    }
}

// ---------------------------------------------------------------------------
extern "C" void kernel_launch(void* const* d_in, const int* in_sizes, int n_in,
                              void* d_out, int out_size, void* d_ws, size_t ws_size,
                              hipStream_t stream) {
    const float* u = (const float*)d_in[0];
    const float* A = (const float*)d_in[1];
    const float* B = (const float*)d_in[2];
    const float* C = (const float*)d_in[3];
    const float* D = (const float*)d_in[4];
    float* y = (float*)d_out;
    float* W = (float*)d_ws;   // 128*128 floats of scratch

    s4_precompute_W<<<1, 256, 2 * S * S * sizeof(float), stream>>>(A, B, C, D, W);

    const int M = in_sizes[0] / S;          // 262144 rows
    const int grid = M / 128;               // 2048 blocks
    s4_gemm_wmma<<<grid, 256, 2 * S * LDS_STRIDE * sizeof(float), stream>>>(u, W, y);
}